// PlaneGrid_16295105921033
// MI455X (gfx1250) — compile-verified
//
#include <hip/hip_runtime.h>
#include <cstdint>
#include <cstddef>

typedef float f4 __attribute__((ext_vector_type(4)));
typedef int v4i __attribute__((vector_size(16)));
typedef __attribute__((address_space(3))) v4i* lds_v4i_ptr;

#define GRID_N   512
#define RCH      16
#define PLANE_EL (512u * 512u * 16u)   /* elems per transposed plane */
#define LDSPITCH 516                   /* 512 + 4 pad: conflict-free transpose */

#if defined(__gfx1250__) && __has_builtin(__builtin_amdgcn_global_load_async_to_lds_b128)
#define HAVE_ASYNC_LDS 1
#else
#define HAVE_ASYNC_LDS 0
#endif

struct Bilerp { int i00, i01, i10, i11; float w00, w01, w10, w11; };

__device__ __forceinline__ Bilerp make_bilerp(float u, float v) {
  Bilerp b;
  float h = (u + 1.0f) * (0.5f * (GRID_N - 1));
  float w = (v + 1.0f) * (0.5f * (GRID_N - 1));
  float h0f = floorf(h), w0f = floorf(w);
  float dh = h - h0f, dw = w - w0f;
  int h0 = (int)h0f, w0 = (int)w0f;
  int h1 = h0 + 1, w1 = w0 + 1;
  float vh0 = (h0 >= 0 && h0 <= GRID_N - 1) ? 1.0f : 0.0f;
  float vh1 = (h1 >= 0 && h1 <= GRID_N - 1) ? 1.0f : 0.0f;
  float vw0 = (w0 >= 0 && w0 <= GRID_N - 1) ? 1.0f : 0.0f;
  float vw1 = (w1 >= 0 && w1 <= GRID_N - 1) ? 1.0f : 0.0f;
  int h0c = min(max(h0, 0), GRID_N - 1), h1c = min(max(h1, 0), GRID_N - 1);
  int w0c = min(max(w0, 0), GRID_N - 1), w1c = min(max(w1, 0), GRID_N - 1);
  b.w00 = (1.0f - dh) * (1.0f - dw) * vh0 * vw0;
  b.w01 = (1.0f - dh) * dw * vh0 * vw1;
  b.w10 = dh * (1.0f - dw) * vh1 * vw0;
  b.w11 = dh * dw * vh1 * vw1;
  b.i00 = (h0c << 9) + w0c;
  b.i01 = (h0c << 9) + w1c;
  b.i10 = (h1c << 9) + w0c;
  b.i11 = (h1c << 9) + w1c;
  return b;
}

__device__ __forceinline__ f4 f4_mul(f4 a, float s) {
  f4 r; r.x = a.x * s; r.y = a.y * s; r.z = a.z * s; r.w = a.w * s; return r;
}
__device__ __forceinline__ f4 f4_fma(f4 a, float s, f4 c) {
  f4 r;
  r.x = fmaf(a.x, s, c.x); r.y = fmaf(a.y, s, c.y);
  r.z = fmaf(a.z, s, c.z); r.w = fmaf(a.w, s, c.w);
  return r;
}

// Sample one transposed plane ([H][W][16] channel-last): 4 corners = 4 cachelines.
__device__ __forceinline__ void sample_tp(const float* __restrict__ pl,
                                          const Bilerp& b, f4* __restrict__ acc) {
  const f4* c00 = (const f4*)(pl + (size_t)b.i00 * RCH);
  const f4* c01 = (const f4*)(pl + (size_t)b.i01 * RCH);
  const f4* c10 = (const f4*)(pl + (size_t)b.i10 * RCH);
  const f4* c11 = (const f4*)(pl + (size_t)b.i11 * RCH);
#pragma unroll
  for (int j = 0; j < 4; ++j) {
    f4 a = c00[j], e = c01[j], c = c10[j], d = c11[j];
    f4 r = f4_mul(a, b.w00);
    r = f4_fma(e, b.w01, r);
    r = f4_fma(c, b.w10, r);
    r = f4_fma(d, b.w11, r);
    acc[j] = r;
  }
}

// ---------------------------------------------------------------------------
// Phase 1: transpose [16][512][512] -> [512][512][16], one (plane,row) / block.
// Global->LDS via CDNA5 async-to-LDS (ASYNCcnt path), then conflict-free LDS
// reads and fully contiguous 4KB/block stores.
// ---------------------------------------------------------------------------
__global__ __launch_bounds__(256) void PlaneGrid_transpose_kernel(
    const float* __restrict__ p0, const float* __restrict__ p1,
    const float* __restrict__ p2, float* __restrict__ outp) {
  __shared__ float lds[RCH * LDSPITCH];
  const int h  = blockIdx.x;   // 0..511
  const int pl = blockIdx.y;   // 0..2
  const float* in = (pl == 0) ? p0 : (pl == 1) ? p1 : p2;
  float* out = outp + ((size_t)pl * GRID_N + h) * (GRID_N * RCH);
  const int t = threadIdx.x;

#pragma unroll
  for (int i = 0; i < 8; ++i) {
    int idx4 = t + 256 * i;          // 2048 float4 = 16 c-rows x 512 w
    int c = idx4 >> 7;               // channel row 0..15
    int w = (idx4 & 127) << 2;       // w offset 0..508 step 4
    const float* src = in + ((size_t)(c << 9) + (size_t)h) * GRID_N + w;
#if HAVE_ASYNC_LDS
    __builtin_amdgcn_global_load_async_to_lds_b128(
        (v4i*)src,
        (lds_v4i_ptr)(&lds[c * LDSPITCH + w]),
        0, 0);
#else
    f4 v = __builtin_nontemporal_load((const f4*)src);  // planes read once here
    *(f4*)(&lds[c * LDSPITCH + w]) = v;
#endif
  }
#if HAVE_ASYNC_LDS
#if __has_builtin(__builtin_amdgcn_s_wait_asynccnt)
  __builtin_amdgcn_s_wait_asynccnt(0);
#else
  asm volatile("s_wait_asynccnt 0" ::: "memory");
#endif
#endif
  __syncthreads();

#pragma unroll
  for (int i = 0; i < 8; ++i) {
    int idx4 = t + 256 * i;
    int w = idx4 >> 2;               // 0..511
    int c = (idx4 & 3) << 2;         // 0,4,8,12
    f4 v;
    v.x = lds[(c + 0) * LDSPITCH + w];
    v.y = lds[(c + 1) * LDSPITCH + w];
    v.z = lds[(c + 2) * LDSPITCH + w];
    v.w = lds[(c + 3) * LDSPITCH + w];
    *(f4*)(out + ((size_t)w << 4) + c) = v;  // fully contiguous 4KB/block store
  }
}

// ---------------------------------------------------------------------------
// Phase 2: one lane per point; 48x global_load_b128 (L2-resident planes),
// NT loads for the read-once xyz stream, 12x non-temporal b128 stores
// (keep the 384MB output stream out of L2).
// ---------------------------------------------------------------------------
__global__ __launch_bounds__(256) void PlaneGrid_gather_tp_kernel(
    const float* __restrict__ xyz, const float* __restrict__ tp,
    const float* __restrict__ mn, const float* __restrict__ mx,
    float* __restrict__ out, int n) {
  int p = blockIdx.x * blockDim.x + threadIdx.x;
  if (p >= n) return;
  float px = __builtin_nontemporal_load(xyz + 3 * p + 0);
  float py = __builtin_nontemporal_load(xyz + 3 * p + 1);
  float pz = __builtin_nontemporal_load(xyz + 3 * p + 2);
  float u = (px - mn[0]) * (2.0f / (mx[0] - mn[0])) - 1.0f;
  float v = (py - mn[1]) * (2.0f / (mx[1] - mn[1])) - 1.0f;
  float w = (pz - mn[2]) * (2.0f / (mx[2] - mn[2])) - 1.0f;

  Bilerp bxy = make_bilerp(u, v);
  Bilerp bxz = make_bilerp(u, w);
  Bilerp byz = make_bilerp(v, w);

  f4 acc[12];
  sample_tp(tp + 0ull * PLANE_EL, bxy, acc + 0);
  sample_tp(tp + 1ull * PLANE_EL, bxz, acc + 4);
  sample_tp(tp + 2ull * PLANE_EL, byz, acc + 8);

  f4* dst = (f4*)(out + (size_t)p * 48);
#pragma unroll
  for (int j = 0; j < 12; ++j) __builtin_nontemporal_store(acc[j], dst + j);
}

// ---------------------------------------------------------------------------
// Fallback (workspace too small): gather directly from [R,H,W] layout.
// ---------------------------------------------------------------------------
__global__ __launch_bounds__(256) void PlaneGrid_gather_direct_kernel(
    const float* __restrict__ xyz, const float* __restrict__ xy,
    const float* __restrict__ xz, const float* __restrict__ yz,
    const float* __restrict__ mn, const float* __restrict__ mx,
    float* __restrict__ out, int n) {
  int p = blockIdx.x * blockDim.x + threadIdx.x;
  if (p >= n) return;
  float u = (xyz[3 * p + 0] - mn[0]) * (2.0f / (mx[0] - mn[0])) - 1.0f;
  float v = (xyz[3 * p + 1] - mn[1]) * (2.0f / (mx[1] - mn[1])) - 1.0f;
  float w = (xyz[3 * p + 2] - mn[2]) * (2.0f / (mx[2] - mn[2])) - 1.0f;
  Bilerp bxy = make_bilerp(u, v);
  Bilerp bxz = make_bilerp(u, w);
  Bilerp byz = make_bilerp(v, w);
  alignas(16) float o[48];
#pragma unroll
  for (int c = 0; c < 16; ++c) {
    const float* b0 = xy + (size_t)c * (GRID_N * GRID_N);
    const float* b1 = xz + (size_t)c * (GRID_N * GRID_N);
    const float* b2 = yz + (size_t)c * (GRID_N * GRID_N);
    o[c] = fmaf(b0[bxy.i11], bxy.w11,
            fmaf(b0[bxy.i10], bxy.w10,
             fmaf(b0[bxy.i01], bxy.w01, b0[bxy.i00] * bxy.w00)));
    o[16 + c] = fmaf(b1[bxz.i11], bxz.w11,
                 fmaf(b1[bxz.i10], bxz.w10,
                  fmaf(b1[bxz.i01], bxz.w01, b1[bxz.i00] * bxz.w00)));
    o[32 + c] = fmaf(b2[byz.i11], byz.w11,
                 fmaf(b2[byz.i10], byz.w10,
                  fmaf(b2[byz.i01], byz.w01, b2[byz.i00] * byz.w00)));
  }
  f4* dst = (f4*)(out + (size_t)p * 48);
  const f4* src = (const f4*)o;
#pragma unroll
  for (int j = 0; j < 12; ++j) __builtin_nontemporal_store(src[j], dst + j);
}

extern "C" void kernel_launch(void* const* d_in, const int* in_sizes, int n_in,
                              void* d_out, int out_size, void* d_ws, size_t ws_size,
                              hipStream_t stream) {
  const float* xyz = (const float*)d_in[0];
  const float* xy  = (const float*)d_in[1];
  const float* xz  = (const float*)d_in[2];
  const float* yz  = (const float*)d_in[3];
  const float* mn  = (const float*)d_in[4];
  const float* mx  = (const float*)d_in[5];
  float* out = (float*)d_out;
  const int n = in_sizes[0] / 3;
  const size_t need = 3ull * PLANE_EL * sizeof(float);  // 48 MB

  if (ws_size >= need) {
    dim3 tgrid(GRID_N, 3);
    PlaneGrid_transpose_kernel<<<tgrid, 256, 0, stream>>>(xy, xz, yz, (float*)d_ws);
    PlaneGrid_gather_tp_kernel<<<(n + 255) / 256, 256, 0, stream>>>(
        xyz, (const float*)d_ws, mn, mx, out, n);
  } else {
    PlaneGrid_gather_direct_kernel<<<(n + 255) / 256, 256, 0, stream>>>(
        xyz, xy, xz, yz, mn, mx, out, n);
  }
}